// MinimalRNNCell_56684978372775
// MI455X (gfx1250) — compile-verified
//
#include <hip/hip_runtime.h>

// MinimalRNNCell recurrence on MI455X (gfx1250):
//   per step: h = (b2 + h)[32x32] @ (b + x_t)[32x32], per batch element.
// One wave32 per batch chain; fp32 WMMA 16x16x4; async global->LDS ring for x.
// LDS holds A = h + b2 directly; b and b2 live in registers as WMMA fragments.

#define SIDE   32
#define UNITS  1024
#define BATCH  128
#define TSTEPS 512
#define DEPTH  8              // x ring-buffer depth (steps in flight)
#define RSTR   36             // padded LDS row stride in floats (144B: 16B aligned, no bank conflicts)
#define MATF   (SIDE * RSTR)  // floats per padded 32x32 matrix in LDS

typedef float v2f __attribute__((ext_vector_type(2)));
typedef float v8f __attribute__((ext_vector_type(8)));
typedef int   v4i_ __attribute__((vector_size(16)));

#define HAVE_WMMA_F32 __has_builtin(__builtin_amdgcn_wmma_f32_16x16x4_f32)
#define HAVE_ASYNC    __has_builtin(__builtin_amdgcn_global_load_async_to_lds_b128)

#if __has_builtin(__builtin_amdgcn_s_wait_asynccnt)
#define WAIT_ASYNC(n) __builtin_amdgcn_s_wait_asynccnt(n)
#else
#define WAIT_ASYNC(n) asm volatile("s_wait_asynccnt %0" ::"i"(n) : "memory")
#endif

__device__ __forceinline__ void async_copy_b128(const float* g, float* l) {
  float* gnc = const_cast<float*>(g);
#if HAVE_ASYNC
  __builtin_amdgcn_global_load_async_to_lds_b128(
      (__attribute__((address_space(1))) v4i_*)gnc,
      (__attribute__((address_space(3))) v4i_*)l, 0, 0);
#else
  unsigned lds_addr = (unsigned)(unsigned long long)(__attribute__((address_space(3))) void*)l;
  asm volatile("global_load_async_to_lds_b128 %0, %1, off" ::"v"(lds_addr), "v"(gnc) : "memory");
#endif
}

__global__ __launch_bounds__(32)
void minimal_rnn_chain_kernel(const float* __restrict__ x,
                              const float* __restrict__ bvec,
                              const float* __restrict__ b2vec,
                              const float* __restrict__ h0,
                              float* __restrict__ out) {
  __shared__ float sm[(DEPTH + 3) * MATF];   // ~50 KB
  float* s_x  = sm;                          // ring of DEPTH padded 32x32 x tiles
  float* s_h  = sm + DEPTH * MATF;           // A = h + b2 (row-major, padded)
  float* s_b  = s_h + MATF;                  // b  staging image (init only)
  float* s_b2 = s_b + MATF;                  // b2 staging image (init only)

  const int bId  = blockIdx.x;
  const int lane = threadIdx.x;
  const int lo   = lane & 15;
  const int hi   = lane >> 4;

  // Init: padded LDS images; s_h starts as h0 + b2 (= A for step 0).
  for (int j = lane; j < UNITS; j += 32) {
    const int r = j >> 5, c = j & 31;
    const float b2v = b2vec[j];
    s_b[r * RSTR + c]  = bvec[j];
    s_b2[r * RSTR + c] = b2v;
    s_h[r * RSTR + c]  = h0[bId * UNITS + j] + b2v;
  }

  const float* xb = x + (size_t)bId * TSTEPS * UNITS;

  // Prologue: fill the async ring (8 x b128 per step; 512B/wave per instr).
  for (int d = 0; d < DEPTH; ++d) {
    const float* g = xb + (size_t)d * UNITS;
    float* l = s_x + (d & (DEPTH - 1)) * MATF;
#pragma unroll
    for (int i = 0; i < 8; ++i) {
      const int chunk = i * 32 + lane;                 // 16B chunk id within 4KB tile
      async_copy_b128(g + chunk * 4,
                      l + (chunk >> 3) * RSTR + (chunk & 7) * 4);
    }
  }

  __syncthreads();

  // Hoist b into B-layout fragments and b2 into C/D-layout fragments (registers).
  v2f  bB[2][8];
  float b2cd[2][2][8];
#pragma unroll
  for (int m = 0; m < 8; ++m) {
    const int kb = m * 4 + 2 * hi;
#pragma unroll
    for (int u = 0; u < 2; ++u) {
      const int ob = kb * RSTR + u * 16 + lo;
      bB[u][m].x = s_b[ob];
      bB[u][m].y = s_b[ob + RSTR];
    }
  }
#pragma unroll
  for (int ti = 0; ti < 2; ++ti)
#pragma unroll
    for (int tj = 0; tj < 2; ++tj)
#pragma unroll
      for (int r = 0; r < 8; ++r)
        b2cd[ti][tj][r] = s_b2[(ti * 16 + 8 * hi + r) * RSTR + tj * 16 + lo];

  v8f acc[2][2];

  for (int t = 0; t < TSTEPS; ++t) {
    // Keep DEPTH-1 future steps in flight.
    if (t + DEPTH < TSTEPS) {
      const float* g = xb + (size_t)(t + DEPTH) * UNITS;
      float* l = s_x + ((t + DEPTH) & (DEPTH - 1)) * MATF;
#pragma unroll
      for (int i = 0; i < 8; ++i) {
        const int chunk = i * 32 + lane;
        async_copy_b128(g + chunk * 4,
                        l + (chunk >> 3) * RSTR + (chunk & 7) * 4);
      }
    }
    // Async loads complete in order: <= 8*(DEPTH-1) outstanding => slot t landed.
    WAIT_ASYNC(8 * (DEPTH - 1));

    const float* xs = s_x + (t & (DEPTH - 1)) * MATF;

    // Gather WMMA fragments for the full K=32 contraction (8 chunks of 4).
    // A frag (16x4 f32): lane = row M=lo, regs = K kb,kb+1 (kb = 4m + 2*hi); pure b64 loads.
    // B frag (4x16 f32): lane = col N=lo, regs = rows kb,kb+1; x gather + register b.
    v2f aF[2][8], bF[2][8];
#pragma unroll
    for (int m = 0; m < 8; ++m) {
      const int kb = m * 4 + 2 * hi;
#pragma unroll
      for (int u = 0; u < 2; ++u) {
        const int oa = (u * 16 + lo) * RSTR + kb;      // A tile ti=u
        aF[u][m].x = s_h[oa];
        aF[u][m].y = s_h[oa + 1];
        const int ob = kb * RSTR + u * 16 + lo;        // B tile tj=u
        bF[u][m].x = xs[ob]        + bB[u][m].x;
        bF[u][m].y = xs[ob + RSTR] + bB[u][m].y;
      }
    }

    // 2x2 output tiles, 8 chained WMMAs each (32 v_wmma per step).
#pragma unroll
    for (int ti = 0; ti < 2; ++ti) {
#pragma unroll
      for (int tj = 0; tj < 2; ++tj) {
        v8f c = {0.f, 0.f, 0.f, 0.f, 0.f, 0.f, 0.f, 0.f};
#pragma unroll
        for (int m = 0; m < 8; ++m) {
#if HAVE_WMMA_F32
          c = __builtin_amdgcn_wmma_f32_16x16x4_f32(
              false, aF[ti][m], false, bF[tj][m], (short)0, c, false, false);
#else
          // Diagnostic fallback only (histogram will show wmma=0).
#pragma unroll
          for (int r = 0; r < 8; ++r)
            c[r] += aF[ti][m].x * bF[tj][m].x + aF[ti][m].y * bF[tj][m].y;
#endif
        }
        acc[ti][tj] = c;
      }
    }

    // Scatter next-step A = h_new + b2 back to LDS (C/D reg r -> row ti*16 + 8*hi + r).
    // Single wave: DS ops complete in order, so the aF reads above are safe.
#pragma unroll
    for (int ti = 0; ti < 2; ++ti)
#pragma unroll
      for (int tj = 0; tj < 2; ++tj)
#pragma unroll
        for (int r = 0; r < 8; ++r)
          s_h[(ti * 16 + 8 * hi + r) * RSTR + tj * 16 + lo] =
              acc[ti][tj][r] + b2cd[ti][tj][r];
  }

  // Final hidden state (raw h, straight from accumulators) -> global.
#pragma unroll
  for (int ti = 0; ti < 2; ++ti)
#pragma unroll
    for (int tj = 0; tj < 2; ++tj)
#pragma unroll
      for (int r = 0; r < 8; ++r)
        out[bId * UNITS + (ti * 16 + 8 * hi + r) * SIDE + tj * 16 + lo] =
            acc[ti][tj][r];
}

extern "C" void kernel_launch(void* const* d_in, const int* in_sizes, int n_in,
                              void* d_out, int out_size, void* d_ws, size_t ws_size,
                              hipStream_t stream) {
  (void)in_sizes; (void)n_in; (void)out_size; (void)d_ws; (void)ws_size;
  const float* x  = (const float*)d_in[0];
  const float* b  = (const float*)d_in[1];
  const float* b2 = (const float*)d_in[2];
  const float* h0 = (const float*)d_in[3];
  float* out = (float*)d_out;
  minimal_rnn_chain_kernel<<<BATCH, 32, 0, stream>>>(x, b, b2, h0, out);
}